// AdaptiveFNOFilter1d_24318104830066
// MI455X (gfx1250) — compile-verified
//
#include <hip/hip_runtime.h>
#include <hip/hip_bf16.h>
#include <math.h>

// Problem constants (from reference)
#define B_    8
#define L_    4096
#define D_    768
#define NM_   2049            // L/2 + 1
#define NB_   8
#define BS_   96
#define MTOT  (B_ * NM_)      // 16392 rows in the spectral GEMM
#define LAMBDA 0.01f
#define ORTHO  (1.0f / 64.0f) // 1/sqrt(4096)

typedef __attribute__((ext_vector_type(2))) float v2f;
typedef __attribute__((ext_vector_type(8))) float v8f;
typedef __attribute__((ext_vector_type(4))) unsigned int u32x4;
typedef __attribute__((ext_vector_type(8))) int i32x8;
typedef __attribute__((ext_vector_type(4))) int i32x4;

// ---------------------------------------------------------------------------
// TDM: stage a contiguous run of `nelems` f32 from global into LDS using a
// Tensor DMA descriptor (1-D tile: tensor_dim0 = tile_dim0 = nelems).
// D# bit layout per CDNA5 ISA ch.8 (group0: count/lds_addr/global_addr/type,
// group1: data_size/dims/tile/strides). Wave-level op, EXEC ignored,
// completion tracked by TENSORcnt. 6-arg builtin form (clang-23 toolchain).
// ---------------------------------------------------------------------------
__device__ __forceinline__ void tdm_load_f32(const float* gsrc, unsigned lds_byte_off,
                                             unsigned nelems) {
  const unsigned long long ga = (unsigned long long)(uintptr_t)gsrc;
  u32x4 g0;
  g0.x = 1u;                                                 // count=1, user descriptor
  g0.y = lds_byte_off;                                       // lds_addr (bytes)
  g0.z = (unsigned)(ga & 0xffffffffull);                     // global_addr[31:0]
  g0.w = (unsigned)((ga >> 32) & 0x1ffffffull) | (2u << 30); // global_addr[56:32] | type=2
  i32x8 g1;
  g1.s0 = 0x20000;                                  // data_size=2 (4 bytes); no multicast
  g1.s1 = (int)((nelems & 0xffffu) << 16);          // tensor_dim0[15:0]  @ bits 63:48
  g1.s2 = (int)((nelems >> 16) | (1u << 16));       // tensor_dim0[31:16] | tensor_dim1=1
  g1.s3 = (int)((nelems & 0xffffu) << 16);          // tile_dim0 = nelems @ bits 127:112
  g1.s4 = 1;                                        // tile_dim1 = 1, tile_dim2 = 0
  g1.s5 = (int)nelems;                              // tensor_dim0_stride[31:0]
  g1.s6 = 0;                                        // stride hi / dim1_stride
  g1.s7 = 0;
  i32x4 gz4;
  gz4.x = 0; gz4.y = 0; gz4.z = 0; gz4.w = 0;       // groups 2/3 unused (<=2D tensor)
  i32x8 gz8;
  gz8.s0 = 0; gz8.s1 = 0; gz8.s2 = 0; gz8.s3 = 0;
  gz8.s4 = 0; gz8.s5 = 0; gz8.s6 = 0; gz8.s7 = 0;
  __builtin_amdgcn_tensor_load_to_lds(g0, g1, gz4, gz4, gz8, 0);
}

// ---------------------------------------------------------------------------
// Forward rFFT: one workgroup handles channels (d, d+1) of batch b, packed as
// one complex length-4096 Stockham FFT in LDS. Output: interleaved complex
// spectrum spec[b][k][d] for k = 0..2048, ortho-normalized.
// ---------------------------------------------------------------------------
__global__ __launch_bounds__(256) void afno_fwd_fft(const float* __restrict__ x,
                                                    float* __restrict__ spec) {
  __shared__ float2 buf[2][L_];
  const int b   = blockIdx.x / (D_ / 2);
  const int d   = (blockIdx.x % (D_ / 2)) * 2;
  const int tid = threadIdx.x;

  for (int l = tid; l < L_; l += 256) {
    buf[0][l] = *(const float2*)(x + ((size_t)(b * L_ + l)) * D_ + d);
  }
  __syncthreads();

  // Stockham radix-2 DIF, natural-order output, sign = -1 (forward)
  int cur = 0;
  int ls  = 0;
  for (int m = L_; m > 1; m >>= 1, ++ls) {
    const int mh = m >> 1;
    const int s  = 1 << ls;
    const float2* __restrict__ src = buf[cur];
    float2* __restrict__ dst       = buf[cur ^ 1];
    for (int it = 0; it < (L_ / 2) / 256; ++it) {
      const int t = tid + (it << 8);
      const int j = t & (s - 1);
      const int i = t >> ls;
      float sw, cw;
      __sincosf(-6.28318530717958647692f * (float)i / (float)m, &sw, &cw);
      const float2 a = src[j + s * i];
      const float2 c = src[j + s * (i + mh)];
      dst[j + s * (2 * i)] = make_float2(a.x + c.x, a.y + c.y);
      const float dr = a.x - c.x, di = a.y - c.y;
      dst[j + s * (2 * i + 1)] = make_float2(dr * cw - di * sw, dr * sw + di * cw);
    }
    __syncthreads();
    cur ^= 1;
  }
  const float2* __restrict__ Z = buf[cur];

  // Unpack two real spectra: A[k] = (Z[k]+conj(Z[N-k]))/2, B[k] = -i(Z[k]-conj(Z[N-k]))/2
  for (int k = tid; k <= L_ / 2; k += 256) {
    const float2 z  = Z[k];
    const float2 zm = Z[(L_ - k) & (L_ - 1)];
    float4 o;
    o.x = 0.5f * ORTHO * (z.x + zm.x);   // A.re  (channel d)
    o.y = 0.5f * ORTHO * (z.y - zm.y);   // A.im
    o.z = 0.5f * ORTHO * (z.y + zm.y);   // B.re  (channel d+1)
    o.w = 0.5f * ORTHO * (zm.x - z.x);   // B.im
    *(float4*)(spec + 2 * (((size_t)b * NM_ + k) * D_ + d)) = o;
  }
}

// ---------------------------------------------------------------------------
// Spectral block-diagonal complex MLP, in place on spec.
// Grid: (ceil(MTOT/64), NB). Block: 128 threads = 4 waves; each wave owns a
// 16-row tile of the (MTOT x 96) per-block GEMM. Weights staged into LDS by
// the Tensor Data Mover; complex GEMM via V_WMMA_F32_16X16X4_F32 (K=96).
// ---------------------------------------------------------------------------
__global__ __launch_bounds__(128) void afno_block_mlp(float* spec,
                                                      const float* __restrict__ w1,
                                                      const float* __restrict__ b1,
                                                      const float* __restrict__ w2,
                                                      const float* __restrict__ b2) {
  __shared__ __align__(16) float smem[4 * BS_ * BS_ + 4 * BS_];
  float* sW1r = smem;                   // reused as sO1r after layer 1
  float* sW1i = smem + BS_ * BS_;       // reused as sO1i after layer 1
  float* sW2r = smem + 2 * BS_ * BS_;
  float* sW2i = smem + 3 * BS_ * BS_;
  float* sBia = smem + 4 * BS_ * BS_;   // [b1r | b1i | b2r | b2i]
  float* sO1r = sW1r;
  float* sO1i = sW1i;

  const int blk = blockIdx.y;
  const int tid = threadIdx.x;

  // ---- weight staging via Tensor Data Mover (wave 0 issues 4 descriptors) ----
  if (tid < 32) {
    const unsigned ldsBase = (unsigned)(uintptr_t)smem;  // AS3 offset in low 32 bits
    const size_t wb = (size_t)blk * BS_ * BS_;
    tdm_load_f32(w1 + wb,                           ldsBase,                          BS_ * BS_);
    tdm_load_f32(w1 + (size_t)NB_ * BS_ * BS_ + wb, ldsBase + 1 * BS_ * BS_ * 4,     BS_ * BS_);
    tdm_load_f32(w2 + wb,                           ldsBase + 2 * BS_ * BS_ * 4,     BS_ * BS_);
    tdm_load_f32(w2 + (size_t)NB_ * BS_ * BS_ + wb, ldsBase + 3 * BS_ * BS_ * 4,     BS_ * BS_);
    __builtin_amdgcn_s_wait_tensorcnt(0);
  }
  if (tid < BS_) {
    sBia[tid]            = b1[(size_t)blk * BS_ + tid];
    sBia[BS_ + tid]      = b1[(size_t)(NB_ + blk) * BS_ + tid];
    sBia[2 * BS_ + tid]  = b2[(size_t)blk * BS_ + tid];
    sBia[3 * BS_ + tid]  = b2[(size_t)(NB_ + blk) * BS_ + tid];
  }
  __syncthreads();

  const int lane    = tid & 31;
  const int wv      = tid >> 5;       // wave 0..3
  const int rlane   = lane & 15;
  const int hi      = lane >> 4;      // lanes 16..31 supply K = +2,+3 of each step
  const int rowBase = blockIdx.x * 64 + wv * 16;

  int arow = rowBase + rlane;         // A-fragment row (clamped; OOB results never stored)
  if (arow >= MTOT) arow = MTOT - 1;
  const float* Abase = spec + 2 * ((size_t)arow * D_ + blk * BS_);

  // ---- layer 1: o1 = relu(Xr@W1r - Xi@W1i + b1r , Xi@W1r + Xr@W1i + b1i) ----
  v8f c1r[6], c1i[6];
  #pragma unroll
  for (int t = 0; t < 6; ++t)
    #pragma unroll
    for (int v = 0; v < 8; ++v) { c1r[t][v] = 0.0f; c1i[t][v] = 0.0f; }

  for (int kk = 0; kk < 24; ++kk) {
    const int k = kk * 4 + hi * 2;
    const float4 xv = *(const float4*)(Abase + 2 * k);  // re[k],im[k],re[k+1],im[k+1]
    v2f ar;  ar.x  = xv.x;  ar.y  = xv.z;
    v2f ai;  ai.x  = xv.y;  ai.y  = xv.w;
    v2f nai; nai.x = -xv.y; nai.y = -xv.w;
    #pragma unroll
    for (int t = 0; t < 6; ++t) {
      const int n = t * 16 + rlane;
      v2f br; br.x = sW1r[k * BS_ + n]; br.y = sW1r[(k + 1) * BS_ + n];
      v2f bi; bi.x = sW1i[k * BS_ + n]; bi.y = sW1i[(k + 1) * BS_ + n];
      c1r[t] = __builtin_amdgcn_wmma_f32_16x16x4_f32(false, ar,  false, br, (short)0, c1r[t], false, false);
      c1r[t] = __builtin_amdgcn_wmma_f32_16x16x4_f32(false, nai, false, bi, (short)0, c1r[t], false, false);
      c1i[t] = __builtin_amdgcn_wmma_f32_16x16x4_f32(false, ai,  false, br, (short)0, c1i[t], false, false);
      c1i[t] = __builtin_amdgcn_wmma_f32_16x16x4_f32(false, ar,  false, bi, (short)0, c1i[t], false, false);
    }
  }

  __syncthreads();  // everyone done reading w1 LDS before we overwrite it with o1

  // bias + relu, write o1 tile to LDS (C layout -> row-major re-staging for layer-2 A)
  #pragma unroll
  for (int t = 0; t < 6; ++t) {
    const int n = t * 16 + rlane;
    const float bre = sBia[n];
    const float bim = sBia[BS_ + n];
    #pragma unroll
    for (int v = 0; v < 8; ++v) {
      const int m = v + 8 * hi;  // C/D layout: lanes 16-31 hold rows 8..15
      float r  = c1r[t][v] + bre; r  = r  > 0.f ? r  : 0.f;
      float im = c1i[t][v] + bim; im = im > 0.f ? im : 0.f;
      sO1r[(wv * 16 + m) * BS_ + n] = r;
      sO1i[(wv * 16 + m) * BS_ + n] = im;
    }
  }
  __syncthreads();

  // ---- layer 2: o2 = o1 (complex) @ W2 + b2, then softshrink ----
  v8f c2r[6], c2i[6];
  #pragma unroll
  for (int t = 0; t < 6; ++t)
    #pragma unroll
    for (int v = 0; v < 8; ++v) { c2r[t][v] = 0.0f; c2i[t][v] = 0.0f; }

  const int rr = (wv * 16 + rlane) * BS_;
  for (int kk = 0; kk < 24; ++kk) {
    const int k = kk * 4 + hi * 2;
    v2f ar;  ar.x  = sO1r[rr + k];  ar.y  = sO1r[rr + k + 1];
    v2f ai;  ai.x  = sO1i[rr + k];  ai.y  = sO1i[rr + k + 1];
    v2f nai; nai.x = -ai.x;         nai.y = -ai.y;
    #pragma unroll
    for (int t = 0; t < 6; ++t) {
      const int n = t * 16 + rlane;
      v2f br; br.x = sW2r[k * BS_ + n]; br.y = sW2r[(k + 1) * BS_ + n];
      v2f bi; bi.x = sW2i[k * BS_ + n]; bi.y = sW2i[(k + 1) * BS_ + n];
      c2r[t] = __builtin_amdgcn_wmma_f32_16x16x4_f32(false, ar,  false, br, (short)0, c2r[t], false, false);
      c2r[t] = __builtin_amdgcn_wmma_f32_16x16x4_f32(false, nai, false, bi, (short)0, c2r[t], false, false);
      c2i[t] = __builtin_amdgcn_wmma_f32_16x16x4_f32(false, ai,  false, br, (short)0, c2i[t], false, false);
      c2i[t] = __builtin_amdgcn_wmma_f32_16x16x4_f32(false, ar,  false, bi, (short)0, c2i[t], false, false);
    }
  }

  // bias + softshrink + predicated in-place store
  #pragma unroll
  for (int t = 0; t < 6; ++t) {
    const int n = t * 16 + rlane;
    const float bre = sBia[2 * BS_ + n];
    const float bim = sBia[3 * BS_ + n];
    #pragma unroll
    for (int v = 0; v < 8; ++v) {
      const int row = rowBase + v + 8 * hi;
      if (row < MTOT) {
        float r  = c2r[t][v] + bre;
        float im = c2i[t][v] + bim;
        float tr = fabsf(r)  - LAMBDA; r  = tr > 0.f ? copysignf(tr, r)  : 0.f;
        float ti = fabsf(im) - LAMBDA; im = ti > 0.f ? copysignf(ti, im) : 0.f;
        *(float2*)(spec + 2 * ((size_t)row * D_ + blk * BS_ + n)) = make_float2(r, im);
      }
    }
  }
}

// ---------------------------------------------------------------------------
// Inverse rFFT + residual. Rebuild full Hermitian spectrum for channel pair
// (d gets Re, d+1 gets Im of the packed inverse), inverse Stockham, add x.
// ---------------------------------------------------------------------------
__global__ __launch_bounds__(256) void afno_inv_fft(const float* __restrict__ spec,
                                                    const float* __restrict__ x,
                                                    float* __restrict__ y) {
  __shared__ float2 buf[2][L_];
  const int b   = blockIdx.x / (D_ / 2);
  const int d   = (blockIdx.x % (D_ / 2)) * 2;
  const int tid = threadIdx.x;

  for (int k = tid; k <= L_ / 2; k += 256) {
    float4 v = *(const float4*)(spec + 2 * (((size_t)b * NM_ + k) * D_ + d));
    if (k == 0 || k == L_ / 2) { v.y = 0.f; v.w = 0.f; }  // irfft ignores DC/Nyquist imag
    buf[0][k] = make_float2(v.x - v.w, v.y + v.z);        // Xa + i*Xb
    if (k != 0 && k != L_ / 2)
      buf[0][L_ - k] = make_float2(v.x + v.w, v.z - v.y); // conj(Xa) + i*conj(Xb)
  }
  __syncthreads();

  int cur = 0;
  int ls  = 0;
  for (int m = L_; m > 1; m >>= 1, ++ls) {
    const int mh = m >> 1;
    const int s  = 1 << ls;
    const float2* __restrict__ src = buf[cur];
    float2* __restrict__ dst       = buf[cur ^ 1];
    for (int it = 0; it < (L_ / 2) / 256; ++it) {
      const int t = tid + (it << 8);
      const int j = t & (s - 1);
      const int i = t >> ls;
      float sw, cw;
      __sincosf(6.28318530717958647692f * (float)i / (float)m, &sw, &cw);
      const float2 a = src[j + s * i];
      const float2 c = src[j + s * (i + mh)];
      dst[j + s * (2 * i)] = make_float2(a.x + c.x, a.y + c.y);
      const float dr = a.x - c.x, di = a.y - c.y;
      dst[j + s * (2 * i + 1)] = make_float2(dr * cw - di * sw, dr * sw + di * cw);
    }
    __syncthreads();
    cur ^= 1;
  }
  const float2* __restrict__ Z = buf[cur];

  for (int l = tid; l < L_; l += 256) {
    const size_t off = ((size_t)(b * L_ + l)) * D_ + d;
    const float2 r = *(const float2*)(x + off);
    const float2 z = Z[l];
    *(float2*)(y + off) = make_float2(z.x * ORTHO + r.x, z.y * ORTHO + r.y);
  }
}

// ---------------------------------------------------------------------------
extern "C" void kernel_launch(void* const* d_in, const int* in_sizes, int n_in,
                              void* d_out, int out_size, void* d_ws, size_t ws_size,
                              hipStream_t stream) {
  const float* x  = (const float*)d_in[0];
  const float* w1 = (const float*)d_in[1];
  const float* b1 = (const float*)d_in[2];
  const float* w2 = (const float*)d_in[3];
  const float* b2 = (const float*)d_in[4];
  float* out  = (float*)d_out;
  float* spec = (float*)d_ws;  // (B, NM, D) interleaved complex, ~96 MiB

  const dim3 fftGrid(B_ * (D_ / 2));
  afno_fwd_fft<<<fftGrid, 256, 0, stream>>>(x, spec);

  const dim3 mlpGrid((MTOT + 63) / 64, NB_);
  afno_block_mlp<<<mlpGrid, 128, 0, stream>>>(spec, w1, b1, w2, b2);

  afno_inv_fft<<<fftGrid, 256, 0, stream>>>(spec, x, out);
}